// DefNorConv2d_76321568850099
// MI455X (gfx1250) — compile-verified
//
#include <hip/hip_runtime.h>
#include <math.h>

typedef __attribute__((ext_vector_type(2))) float v2f;
typedef __attribute__((ext_vector_type(8))) float v8f;

#define BATCH 4
#define CIN   64
#define COUT  64
#define H     128
#define W     128
#define HW    (H*W)
#define KK    9            // 3x3 taps
#define CK    (CIN*KK)     // 576 = GEMM K
#define NOFF  (2*KK)       // 18 offset channels
#define XCH   (CIN+1)      // 65 input channels for the small convs
#define CHB   (CIN*4)      // bytes per NHWC pixel row (256)
#define PXB   128          // pixels per block in the WMMA kernel

// ---------------------------------------------------------------------------
// Kernel 0a: permute weight (Co,Ci,3,3) -> Wp[co][k*64+ci] (tap-major K).
// ---------------------------------------------------------------------------
__global__ void __launch_bounds__(256)
permute_weight_kernel(const float* __restrict__ w, float* __restrict__ wp)
{
    int t = blockIdx.x * blockDim.x + threadIdx.x;       // < COUT*CK
    int co = t / CK;
    int r  = t - co * CK;
    int k  = r >> 6;
    int ci = r & 63;
    wp[t] = w[co * CK + ci * KK + k];
}

// ---------------------------------------------------------------------------
// Kernel 0b: transpose x NCHW -> NHWC (xT[b][p][ci]) with an LDS tile.
// ---------------------------------------------------------------------------
__global__ void __launch_bounds__(256)
nchw_to_nhwc_kernel(const float* __restrict__ x, float* __restrict__ xT)
{
    __shared__ float tile[CIN][65];          // +1 pad: kill bank conflicts

    const int tilesPerBatch = HW / 64;       // 256
    int b    = blockIdx.x / tilesPerBatch;
    int base = (blockIdx.x - b * tilesPerBatch) * 64;

    #pragma unroll
    for (int i = 0; i < 16; ++i) {
        int idx  = threadIdx.x + i * 256;    // < 4096
        int c    = idx >> 6;
        int psub = idx & 63;
        tile[c][psub] = x[((size_t)b * CIN + c) * HW + base + psub];
    }
    __syncthreads();
    #pragma unroll
    for (int i = 0; i < 16; ++i) {
        int idx  = threadIdx.x + i * 256;
        int c    = idx & 63;                 // fastest in lane -> coalesced write
        int psub = idx >> 6;
        xT[((size_t)b * HW + base + psub) * CIN + c] = tile[c][psub];
    }
}

// ---------------------------------------------------------------------------
// Kernel 1: fused 3x3 conv over concat(x, mask) -> offsets (18ch, raw) and
// dcn_mask (9ch, sigmoid). Branchless taps: clamp + multiply by in-bounds.
// ---------------------------------------------------------------------------
__global__ void __launch_bounds__(256)
conv_offset_mask_kernel(const float* __restrict__ x,
                        const float* __restrict__ mask_in,
                        const float* __restrict__ offset_w,
                        const float* __restrict__ offset_b,
                        const float* __restrict__ mask_w,
                        const float* __restrict__ mask_b,
                        float* __restrict__ offs,
                        float* __restrict__ dcnm)
{
    int t = blockIdx.x * blockDim.x + threadIdx.x;   // 0 .. B*HW-1
    int b = t / HW;
    int p = t - b * HW;
    int h = p >> 7, w = p & 127;

    float acc[NOFF + KK];
    #pragma unroll
    for (int c = 0; c < NOFF; ++c) acc[c] = offset_b[c];
    #pragma unroll
    for (int c = 0; c < KK; ++c) acc[NOFF + c] = mask_b[c];

    for (int ci = 0; ci < XCH; ++ci) {
        const float* plane = (ci < CIN) ? (x + ((size_t)b * CIN + ci) * HW)
                                        : (mask_in + (size_t)b * HW);
        #pragma unroll
        for (int ky = 0; ky < 3; ++ky) {
            int yy  = h + ky - 1;
            int yyc = min(max(yy, 0), H - 1);
            #pragma unroll
            for (int kx = 0; kx < 3; ++kx) {
                int xx  = w + kx - 1;
                int xxc = min(max(xx, 0), W - 1);
                float inb = (yy == yyc && xx == xxc) ? 1.0f : 0.0f;
                float v = plane[yyc * W + xxc] * inb;
                int wi = (ci * 3 + ky) * 3 + kx;
                #pragma unroll
                for (int c = 0; c < NOFF; ++c)
                    acc[c] += v * offset_w[c * (XCH * KK) + wi];
                #pragma unroll
                for (int c = 0; c < KK; ++c)
                    acc[NOFF + c] += v * mask_w[c * (XCH * KK) + wi];
            }
        }
    }
    #pragma unroll
    for (int c = 0; c < NOFF; ++c)
        offs[((size_t)b * NOFF + c) * HW + p] = acc[c];
    #pragma unroll
    for (int c = 0; c < KK; ++c)
        dcnm[((size_t)b * KK + c) * HW + p] =
            1.0f / (1.0f + __expf(-acc[NOFF + c]));
}

// ---------------------------------------------------------------------------
// Kernel 2: update_mask = clip(64 * sum_k bilinear(mask, pos_k), 0, 1)
// ---------------------------------------------------------------------------
__global__ void __launch_bounds__(256)
update_mask_kernel(const float* __restrict__ mask_in,
                   const float* __restrict__ offs,
                   float* __restrict__ umask)
{
    int t = blockIdx.x * blockDim.x + threadIdx.x;
    int b = t / HW;
    int p = t - b * HW;
    int h = p >> 7, w = p & 127;
    const float* plane = mask_in + (size_t)b * HW;
    float s = 0.0f;
    #pragma unroll
    for (int k = 0; k < KK; ++k) {
        float dy = offs[((size_t)b * NOFF + 2 * k    ) * HW + p];
        float dx = offs[((size_t)b * NOFF + 2 * k + 1) * HW + p];
        float py = dy + (float)(k / 3) + (float)(h - 1);
        float px = dx + (float)(k % 3) + (float)(w - 1);
        float fy = floorf(py), fx = floorf(px);
        int y0 = (int)fy, x0 = (int)fx;
        float wy = py - fy, wx = px - fx;
        int y1 = y0 + 1, x1 = x0 + 1;
        int y0c = min(max(y0, 0), H - 1), x0c = min(max(x0, 0), W - 1);
        int y1c = min(max(y1, 0), H - 1), x1c = min(max(x1, 0), W - 1);
        float vy0 = (y0 == y0c) ? 1.0f : 0.0f, vx0 = (x0 == x0c) ? 1.0f : 0.0f;
        float vy1 = (y1 == y1c) ? 1.0f : 0.0f, vx1 = (x1 == x1c) ? 1.0f : 0.0f;
        s += plane[y0c * W + x0c] * ((1.0f - wy) * (1.0f - wx) * vy0 * vx0)
           + plane[y0c * W + x1c] * ((1.0f - wy) * wx          * vy0 * vx1)
           + plane[y1c * W + x0c] * (wy * (1.0f - wx)          * vy1 * vx0)
           + plane[y1c * W + x1c] * (wy * wx                   * vy1 * vx1);
    }
    s *= (float)CIN;
    s = fminf(fmaxf(s, 0.0f), 1.0f);
    umask[(size_t)b * HW + p] = s;
}

// ---------------------------------------------------------------------------
// Kernel 3: deformable implicit GEMM via V_WMMA_F32_16X16X4_F32 over NHWC x.
// Block = 128 threads (4 waves), 128 pixels/block. Each wave owns TWO
// 16-pixel N tiles and all four M=16 tiles, so each A fragment (weights)
// feeds two WMMAs: per K-step = 8 corner b64 loads + 4 weight b64 loads
// + 8 WMMAs (1.5 loads/WMMA). Corner reads are b64 at base + 32-bit VGPR
// offset with the channel offset folded into the immediate.
// ---------------------------------------------------------------------------
__global__ void __launch_bounds__(128)
deform_wmma_kernel(const float* __restrict__ xT,    // NHWC
                   const float* __restrict__ wp,    // permuted weight
                   const float* __restrict__ bias,
                   const float* __restrict__ offs,
                   const float* __restrict__ dcnm,
                   const float* __restrict__ umask,
                   float* __restrict__ out)
{
    __shared__ uint4  s_off[PXB * KK];
    __shared__ float4 s_wt [PXB * KK];

    const int tilesPerBatch = HW / PXB;                // 128
    int b    = blockIdx.x / tilesPerBatch;
    int base = (blockIdx.x - b * tilesPerBatch) * PXB; // pixel base

    // Phase 1: sampling metadata for 128 pixels x 9 taps (branchless)
    for (int ent = threadIdx.x; ent < PXB * KK; ent += blockDim.x) {
        int nl = ent / KK, k = ent - nl * KK;
        int p  = base + nl;
        int h  = p >> 7, w = p & 127;
        float dy = offs[((size_t)b * NOFF + 2 * k    ) * HW + p];
        float dx = offs[((size_t)b * NOFF + 2 * k + 1) * HW + p];
        float py = dy + (float)(k / 3) + (float)(h - 1);
        float px = dx + (float)(k % 3) + (float)(w - 1);
        float fy = floorf(py), fx = floorf(px);
        int y0 = (int)fy, x0 = (int)fx;
        float wy = py - fy, wx = px - fx;
        int y1 = y0 + 1, x1 = x0 + 1;
        int y0c = min(max(y0, 0), H - 1), x0c = min(max(x0, 0), W - 1);
        int y1c = min(max(y1, 0), H - 1), x1c = min(max(x1, 0), W - 1);
        float vy0 = (y0 == y0c) ? 1.0f : 0.0f, vx0 = (x0 == x0c) ? 1.0f : 0.0f;
        float vy1 = (y1 == y1c) ? 1.0f : 0.0f, vx1 = (x1 == x1c) ? 1.0f : 0.0f;
        float m = dcnm[((size_t)b * KK + k) * HW + p];
        s_off[ent] = make_uint4((unsigned)(y0c * W + x0c) * CHB,
                                (unsigned)(y0c * W + x1c) * CHB,
                                (unsigned)(y1c * W + x0c) * CHB,
                                (unsigned)(y1c * W + x1c) * CHB);
        s_wt[ent]  = make_float4((1.0f - wy) * (1.0f - wx) * vy0 * vx0 * m,
                                 (1.0f - wy) * wx          * vy0 * vx1 * m,
                                 wy * (1.0f - wx)          * vy1 * vx0 * m,
                                 wy * wx                   * vy1 * vx1 * m);
    }
    __syncthreads();

    int lane   = threadIdx.x & 31;
    int waveId = threadIdx.x >> 5;
    int col    = lane & 15;                 // N within tile / M within tile
    int nlA    = waveId * 32 + col;         // pixel of N tile A
    int nlB    = nlA + 16;                  // pixel of N tile B
    int r0     = (lane < 16) ? 0 : 2;       // ISA 16x4 A / 4x16 B lane split

    const char* xTb = (const char*)(xT + (size_t)b * HW * CIN);
    v8f accA[4] = {v8f{}, v8f{}, v8f{}, v8f{}};
    v8f accB[4] = {v8f{}, v8f{}, v8f{}, v8f{}};

    for (int k = 0; k < KK; ++k) {
        uint4  offA = s_off[nlA * KK + k];  // VGPR-resident for all 16 steps
        float4 wtA  = s_wt [nlA * KK + k];
        uint4  offB = s_off[nlB * KK + k];
        float4 wtB  = s_wt [nlB * KK + k];
        #pragma unroll 4
        for (int ci0 = 0; ci0 < CIN; ci0 += 4) {
            unsigned cb = (unsigned)(ci0 + r0) * 4;   // folds into imm offset
            v2f a00 = *(const v2f*)(xTb + offA.x + cb);
            v2f a01 = *(const v2f*)(xTb + offA.y + cb);
            v2f a10 = *(const v2f*)(xTb + offA.z + cb);
            v2f a11 = *(const v2f*)(xTb + offA.w + cb);
            v2f b00 = *(const v2f*)(xTb + offB.x + cb);
            v2f b01 = *(const v2f*)(xTb + offB.y + cb);
            v2f b10 = *(const v2f*)(xTb + offB.z + cb);
            v2f b11 = *(const v2f*)(xTb + offB.w + cb);
            v2f bfragA = a00 * wtA.x + a01 * wtA.y + a10 * wtA.z + a11 * wtA.w;
            v2f bfragB = b00 * wtB.x + b01 * wtB.y + b10 * wtB.z + b11 * wtB.w;
            int kcol = k * CIN + ci0 + r0;
            #pragma unroll
            for (int m = 0; m < 4; ++m) {
                int co = m * 16 + col;
                const float* wpp = wp + (size_t)co * CK + kcol;
                v2f afrag = { wpp[0], wpp[1] };
                accA[m] = __builtin_amdgcn_wmma_f32_16x16x4_f32(
                    false, afrag, false, bfragA, (short)0, accA[m], false, false);
                accB[m] = __builtin_amdgcn_wmma_f32_16x16x4_f32(
                    false, afrag, false, bfragB, (short)0, accB[m], false, false);
            }
        }
    }

    // Epilogue: bias + update_mask multiply, scatter per C/D lane layout
    int   nA  = base + nlA;
    int   nB  = base + nlB;
    float umA = umask[(size_t)b * HW + nA];
    float umB = umask[(size_t)b * HW + nB];
    #pragma unroll
    for (int m = 0; m < 4; ++m) {
        #pragma unroll
        for (int r = 0; r < 8; ++r) {
            int co = m * 16 + r + ((lane < 16) ? 0 : 8);
            float bb = bias[co];
            out[((size_t)b * COUT + co) * HW + nA] = (accA[m][r] + bb) * umA;
            out[((size_t)b * COUT + co) * HW + nB] = (accB[m][r] + bb) * umB;
        }
    }
}

// ---------------------------------------------------------------------------
extern "C" void kernel_launch(void* const* d_in, const int* in_sizes, int n_in,
                              void* d_out, int out_size, void* d_ws, size_t ws_size,
                              hipStream_t stream)
{
    const float* x        = (const float*)d_in[0];
    const float* mask_in  = (const float*)d_in[1];
    const float* weight   = (const float*)d_in[2];
    const float* bias     = (const float*)d_in[3];
    const float* offset_w = (const float*)d_in[4];
    const float* offset_b = (const float*)d_in[5];
    const float* mask_w   = (const float*)d_in[6];
    const float* mask_b   = (const float*)d_in[7];

    float* outp  = (float*)d_out;
    float* umask = outp + (size_t)BATCH * COUT * HW;     // tuple tail
    float* wsf   = (float*)d_ws;
    float* offs  = wsf;                                  // (B,18,HW)
    float* dcnm  = offs + (size_t)BATCH * NOFF * HW;     // (B,9,HW)
    float* wperm = dcnm + (size_t)BATCH * KK * HW;       // (64,576)
    float* xT    = wperm + (size_t)COUT * CK;            // (B,HW,64) NHWC

    permute_weight_kernel<<<(COUT * CK) / 256, 256, 0, stream>>>(weight, wperm);

    nchw_to_nhwc_kernel<<<(BATCH * HW) / 64, 256, 0, stream>>>(x, xT);

    conv_offset_mask_kernel<<<(BATCH * HW) / 256, 256, 0, stream>>>(
        x, mask_in, offset_w, offset_b, mask_w, mask_b, offs, dcnm);

    update_mask_kernel<<<(BATCH * HW) / 256, 256, 0, stream>>>(
        mask_in, offs, umask);

    deform_wmma_kernel<<<(BATCH * HW) / PXB, 128, 0, stream>>>(
        xT, wperm, bias, offs, dcnm, umask, outp);
}